// TopologyWarpingBCE_11716670783835
// MI455X (gfx1250) — compile-verified
//
#include <hip/hip_runtime.h>

typedef float v2f __attribute__((ext_vector_type(2)));
typedef float v4f __attribute__((ext_vector_type(4)));
typedef float v8f __attribute__((ext_vector_type(8)));

// softplus(p) = max(p,0) + log(1 + exp(-|p|))   (numerically stable)
__device__ __forceinline__ float softplus_f(float p) {
    float t = __builtin_expf(-__builtin_fabsf(p));
    return __builtin_fmaxf(p, 0.0f) + __builtin_logf(1.0f + t);
}

// Deterministic block reduction. Wave stage uses V_WMMA_F32_16X16X4_F32:
//   A = ones(16x4), B holds each lane's partial exactly once (VGPR0=v, VGPR1=0).
//   Since the 64 B-slots are covered bijectively by (lane, vgpr), the matrix sum
//   equals the sum of partials regardless of the exact slot mapping.
//   D[M][N] = colsum[N]; VGPR0 lane l holds colsum[l & 15]  ->  4 xor-shuffles
//   sum the 16 column sums into every lane of each 16-lane half.
// Requires EXEC all-1s: call only at full-block convergence.
__device__ __forceinline__ float block_reduce_wmma(float v, float* lds) {
    v2f a; a.x = 1.0f; a.y = 1.0f;       // ones A-matrix (16x4 f32, 2 VGPRs/lane)
    v2f b; b.x = v;    b.y = 0.0f;       // partial placed once in B (4x16 f32)
    v8f c = {};
    c = __builtin_amdgcn_wmma_f32_16x16x4_f32(
        /*neg_a=*/false, a, /*neg_b=*/false, b,
        /*c_mod=*/(short)0, c, /*reuse_a=*/false, /*reuse_b=*/false);
    float s = c[0];                       // colsum[lane & 15]
    s += __shfl_xor(s, 1, 32);
    s += __shfl_xor(s, 2, 32);
    s += __shfl_xor(s, 4, 32);
    s += __shfl_xor(s, 8, 32);           // wave total (replicated in both halves)

    const int lane = threadIdx.x & 31;
    const int wave = threadIdx.x >> 5;
    if (lane == 0) lds[wave] = s;
    __syncthreads();
    float total = 0.0f;
    if (threadIdx.x == 0) {
        const int nwaves = (blockDim.x + 31) >> 5;
        for (int w = 0; w < nwaves; ++w) total += lds[w];   // fixed order: deterministic
    }
    return total;   // valid in thread 0 only
}

// Pass 1: stream pred/label with B128 non-temporal loads, accumulate
// softplus(p) - l*p per thread, reduce per block, store block partial.
__global__ void __launch_bounds__(256)
bce_partial_kernel(const float* __restrict__ pred,
                   const float* __restrict__ label,
                   float* __restrict__ partial,
                   long long n4, long long n) {
    __shared__ float lds[8];
    const v4f* __restrict__ p4 = (const v4f*)pred;
    const v4f* __restrict__ l4 = (const v4f*)label;

    const long long stride = (long long)gridDim.x * blockDim.x;
    long long i = (long long)blockIdx.x * blockDim.x + threadIdx.x;

    float acc = 0.0f;
#pragma unroll 4
    for (; i < n4; i += stride) {
        v4f p = __builtin_nontemporal_load(p4 + i);   // global_load_b128 th:NT
        v4f l = __builtin_nontemporal_load(l4 + i);
        acc += softplus_f(p.x) - l.x * p.x;
        acc += softplus_f(p.y) - l.y * p.y;
        acc += softplus_f(p.z) - l.z * p.z;
        acc += softplus_f(p.w) - l.w * p.w;
    }
    // Scalar tail (n % 4), handled by block 0 only (<= 3 elements).
    if (blockIdx.x == 0) {
        long long t = 4 * n4 + threadIdx.x;
        if (t < n) {
            float p = pred[t];
            acc += softplus_f(p) - label[t] * p;
        }
    }

    float total = block_reduce_wmma(acc, lds);        // full-block convergence here
    if (threadIdx.x == 0) partial[blockIdx.x] = total;
}

// Pass 2: one block folds the per-block partials in a fixed order.
__global__ void __launch_bounds__(256)
bce_final_kernel(const float* __restrict__ partial,
                 float* __restrict__ out, int nblocks) {
    __shared__ float lds[8];
    float acc = 0.0f;
    for (int i = threadIdx.x; i < nblocks; i += blockDim.x) acc += partial[i];
    float total = block_reduce_wmma(acc, lds);
    if (threadIdx.x == 0) out[0] = total;
}

extern "C" void kernel_launch(void* const* d_in, const int* in_sizes, int n_in,
                              void* d_out, int out_size, void* d_ws, size_t ws_size,
                              hipStream_t stream) {
    const float* pred  = (const float*)d_in[0];
    const float* label = (const float*)d_in[1];
    float* out     = (float*)d_out;
    float* partial = (float*)d_ws;

    const long long n  = (long long)in_sizes[0];   // 33,554,432 elements
    const long long n4 = n >> 2;                   // B128 chunks

    // 4096 blocks x 8 wave32 = 32K waves: saturates 23.3 TB/s HBM; partial
    // buffer is 16 KB of workspace (clamped defensively to ws_size).
    int blocks = 4096;
    if ((size_t)blocks * sizeof(float) > ws_size) {
        blocks = (int)(ws_size / sizeof(float));
        if (blocks < 1) blocks = 1;
    }

    bce_partial_kernel<<<blocks, 256, 0, stream>>>(pred, label, partial, n4, n);
    bce_final_kernel<<<1, 256, 0, stream>>>(partial, out, blocks);
}